// VectorQuantizer_5738076307516
// MI455X (gfx1250) — compile-verified
//
#include <hip/hip_runtime.h>
#include <hip/hip_bf16.h>

typedef __attribute__((ext_vector_type(2))) float v2f;
typedef __attribute__((ext_vector_type(8))) float v8f;

#define N_ROWS   32768          // 8*64*64 query vectors
#define N_CODES  8192
#define TILE     16
#define N_TILES  (N_ROWS / TILE)        // 2048 waves
#define WAVES_PER_BLOCK 8               // 256 threads = 8 wave32
#define ZQ_ELEMS 131072                 // 8*4*64*64
#define LOSS_SCALE (1.25f / 131072.0f)  // (1+BETA) / (N*D)

// One wave per 16-row tile of z_flat. A = -2*z (16x4 f32), loop over the
// codebook in 16-column tiles, C seeded with ||e||^2 so the WMMA directly
// yields the argmin objective  ||e||^2 - 2 z.e  (per-row constant ||z||^2
// dropped -- it cannot change the argmin).
__global__ __launch_bounds__(256)
void vq_argmin_kernel(const float* __restrict__ z,
                      const float* __restrict__ embed,
                      float* __restrict__ out,
                      float* __restrict__ partial)
{
    const int lane = threadIdx.x & 31;
    const int wid  = threadIdx.x >> 5;
    const int tile = blockIdx.x * WAVES_PER_BLOCK + wid;   // 0..2047
    const int base = tile * TILE;

    const int  n  = lane & 15;      // A: row-in-tile, B/C: column-in-tile
    const bool hi = lane >= 16;     // lane half selects K=0,1 vs K=2,3

    // ---- A-matrix (loaded once): lane holds row (base+n), two K components,
    //      pre-scaled by -2.  z is NCHW: offset = ((b*4+c)<<12) + hw.
    int row = base + n;
    int b   = row >> 12;
    int hw  = row & 4095;
    int k0  = hi ? 2 : 0;
    v2f a;
    a.x = -2.0f * z[(((b << 2) + k0    ) << 12) + hw];
    a.y = -2.0f * z[(((b << 2) + k0 + 1) << 12) + hw];

    float best[8];
    int   bidx[8];
    #pragma unroll
    for (int v = 0; v < 8; ++v) { best[v] = 3.4e38f; bidx[v] = 0; }

    const float4* embed4 = (const float4*)embed;   // 16B per codebook row

    for (int j = 0; j < N_CODES / TILE; ++j) {
        int    code = j * TILE + n;
        float4 er   = embed4[code];                // global_load_b128
        float  e2   = er.x*er.x + er.y*er.y + er.z*er.z + er.w*er.w;

        // B-matrix 4x16: VGPR0 = K0(lo)/K2(hi), VGPR1 = K1(lo)/K3(hi)
        v2f bm;
        bm.x = hi ? er.z : er.x;
        bm.y = hi ? er.w : er.y;

        // C seeded with ||e||^2 (column-broadcast: each lane owns one column)
        v8f c;
        #pragma unroll
        for (int v = 0; v < 8; ++v) c[v] = e2;

        // D = (-2z) * e + ||e||^2   (16x16 f32 tile, full precision)
        c = __builtin_amdgcn_wmma_f32_16x16x4_f32(
                /*neg_a=*/false, a, /*neg_b=*/false, bm,
                /*c_mod=*/(short)0, c, /*reuse_a=*/false, /*reuse_b=*/false);

        // running argmin; strict '<' keeps first (lowest) index within a lane
        #pragma unroll
        for (int v = 0; v < 8; ++v) {
            bool p  = c[v] < best[v];
            best[v] = p ? c[v] : best[v];
            bidx[v] = p ? code : bidx[v];
        }
    }

    // ---- reduce across the 16 lanes of each half (columns of the tile).
    // masks 8,4,2,1 stay inside lanes [0,15] / [16,31].
    #pragma unroll
    for (int v = 0; v < 8; ++v) {
        #pragma unroll
        for (int off = 8; off >= 1; off >>= 1) {
            float ov = __shfl_xor(best[v], off, 32);
            int   oi = __shfl_xor(bidx[v], off, 32);
            bool  p  = (ov < best[v]) || (ov == best[v] && oi < bidx[v]);
            best[v] = p ? ov : best[v];
            bidx[v] = p ? oi : bidx[v];
        }
    }

    // ---- epilogue: lane 0 owns rows base+0..7, lane 16 owns base+8..15.
    if (n == 0) {
        int   mbase = hi ? 8 : 0;
        float lsum  = 0.0f;
        #pragma unroll
        for (int v = 0; v < 8; ++v) {
            int r   = base + mbase + v;
            int idx = bidx[v];
            out[ZQ_ELEMS + 1 + r] = (float)idx;          // min_idx (exact < 2^24)
            float4 e  = embed4[idx];                     // gather z_q = embed[idx]
            int rb = r >> 12, rhw = r & 4095;
            float ec[4] = {e.x, e.y, e.z, e.w};
            #pragma unroll
            for (int ch = 0; ch < 4; ++ch) {
                int   off = (((rb << 2) + ch) << 12) + rhw;  // NCHW
                float zc  = z[off];
                out[off]  = ec[ch];                      // z_q_out (STE fwd value)
                float d   = ec[ch] - zc;
                lsum     += d * d;
            }
        }
        partial[tile * 2 + (hi ? 1 : 0)] = lsum;         // deterministic partials
    }
}

// Fixed-order reduction of the 4096 per-half-wave loss partials (no FP
// atomics -> bitwise deterministic across graph replays).
__global__ void vq_loss_reduce(const float* __restrict__ partial,
                               float* __restrict__ out)
{
    if (threadIdx.x == 0 && blockIdx.x == 0) {
        float s = 0.0f;
        for (int i = 0; i < N_TILES * 2; ++i) s += partial[i];
        out[ZQ_ELEMS] = s * LOSS_SCALE;
    }
}

extern "C" void kernel_launch(void* const* d_in, const int* in_sizes, int n_in,
                              void* d_out, int out_size, void* d_ws, size_t ws_size,
                              hipStream_t stream) {
    const float* z     = (const float*)d_in[0];   // [8,4,64,64] f32
    const float* embed = (const float*)d_in[1];   // [8192,4]    f32
    float* out     = (float*)d_out;               // [zq 131072 | loss 1 | idx 32768]
    float* partial = (float*)d_ws;                // 4096 floats of scratch

    (void)in_sizes; (void)n_in; (void)out_size; (void)ws_size;

    vq_argmin_kernel<<<N_TILES / WAVES_PER_BLOCK, WAVES_PER_BLOCK * 32, 0, stream>>>(
        z, embed, out, partial);
    vq_loss_reduce<<<1, 64, 0, stream>>>(partial, out);
}